// MultiParamAttentionInterpolator_24627342475871
// MI455X (gfx1250) — compile-verified
//
#include <hip/hip_runtime.h>
#include <math.h>

#define M_   24
#define TS_  60
#define NX_  192
#define NY_  192
#define H_   4
#define D_   8
#define T_   50
#define NG_  50

typedef __attribute__((ext_vector_type(8))) float v8f;
typedef __attribute__((ext_vector_type(2))) float v2f;

// ---- workspace layout (in 4-byte units) ----
#define W_OFF   0                    // w[24] float
#define TI_OFF  24                   // ti[50*24] int  (t-major)
#define IX_OFF  (TI_OFF + T_*M_)     // ix[50] int
#define FX_OFF  (IX_OFF + NG_)       // fx[50] float
#define MX_OFF  (FX_OFF + NG_)       // mx[50] float
#define IY_OFF  (MX_OFF + NG_)       // iy[24*50] int (m-major)
#define FY_OFF  (IY_OFF + M_*NG_)    // fy[24*50] float
#define MY_OFF  (FY_OFF + M_*NG_)    // my[24*50] float
// total = MY_OFF + 24*50 = 4974 words (~20 KB)

__global__ __launch_bounds__(32)
void setup_kernel(const float* __restrict__ params,
                  const float* __restrict__ src_y,
                  const float* __restrict__ src_t_last,
                  const float* __restrict__ Wq, const float* __restrict__ bq,
                  const float* __restrict__ Wk, const float* __restrict__ bk,
                  const float* __restrict__ ly_target_p,
                  const float* __restrict__ c_cu_p,
                  const float* __restrict__ c_ni_p,
                  float* __restrict__ ws)
{
    const int lane   = threadIdx.x;      // 0..31, single wave
    const int lhalf  = lane & 15;
    const bool hi    = lane >= 16;

    __shared__ float Kmat[32][32];
    __shared__ float Qv[32];
    __shared__ float logits[24][4];
    __shared__ float mh[4], dh[4];
    __shared__ float attn[24], swv[24], wv[24], lysh[24];
    __shared__ float ysc[M_ * NY_];      // 18.4 KB, scaled y grids

    const float ly_target = ly_target_p[0];
    const float t_ly = (ly_target - 30.0f) / 90.0f;
    const float t_cu = c_cu_p[0] / 0.0029f;
    const float t_ni = c_ni_p[0] / 0.0018f;

    // normalized parameter matrix p[m][k], zero-padded to 32 rows / K=4
    auto prow = [&](int r, int k) -> float {
        if (r >= M_ || k >= 3) return 0.0f;
        float v = params[r * 3 + k];
        if (k == 0) return (v - 30.0f) / 90.0f;
        if (k == 1) return v / 0.0029f;
        return v / 0.0018f;
    };

    // ---- K = p @ Wk via V_WMMA_F32_16X16X4_F32 (2 row-tiles x 2 col-tiles) ----
    // A layout (32-bit 16x4): lanes 0-15 -> M=lane, VGPR0=K0, VGPR1=K1;
    //                         lanes 16-31 -> M=lane-16, VGPR0=K2, VGPR1=K3
    // B layout mirrored K-split: lanes 0-15 -> K0/K1, lanes 16-31 -> K2/K3, N=lane%16
    // C/D layout (32-bit 16x16): VGPR g -> (M=g, N=lane) / (M=g+8, N=lane-16)
    for (int rt = 0; rt < 2; ++rt) {
        const int r = rt * 16 + lhalf;
        v2f a;
        a[0] = prow(r, hi ? 2 : 0);
        a[1] = prow(r, hi ? 3 : 1);
        for (int ct = 0; ct < 2; ++ct) {
            const int c = ct * 16 + lhalf;
            v2f b;
            b[0] = hi ? Wk[2 * 32 + c] : Wk[0 * 32 + c];
            b[1] = hi ? 0.0f           : Wk[1 * 32 + c];
            v8f acc = {};
            acc = __builtin_amdgcn_wmma_f32_16x16x4_f32(
                false, a, false, b, (short)0, acc, false, false);
            for (int g = 0; g < 8; ++g) {
                const int row = rt * 16 + g + (hi ? 8 : 0);
                const int col = ct * 16 + lhalf;
                Kmat[row][col] = acc[g] + bk[col];
            }
        }
    }

    // ---- Q = tgt @ Wq + bq (one column per lane) ----
    Qv[lane] = t_ly * Wq[0 * 32 + lane] + t_cu * Wq[1 * 32 + lane]
             + t_ni * Wq[2 * 32 + lane] + bq[lane];
    __syncthreads();

    // ---- logits[m][h] = <K[m,h,:], Q[h,:]> / sqrt(8) ----
    for (int idx = lane; idx < M_ * H_; idx += 32) {
        const int m = idx >> 2, h = idx & 3;
        float s = 0.0f;
        for (int d = 0; d < D_; ++d) s += Kmat[m][h * D_ + d] * Qv[h * D_ + d];
        logits[m][h] = s * 0.35355339059327373f;   // 1/sqrt(8)
    }
    __syncthreads();

    if (lane < H_) {                                  // softmax over m, per head
        const int h = lane;
        float mx = -INFINITY;
        for (int m = 0; m < M_; ++m) mx = fmaxf(mx, logits[m][h]);
        float den = 0.0f;
        for (int m = 0; m < M_; ++m) den += expf(logits[m][h] - mx);
        mh[h] = mx; dh[h] = den;
    }
    __syncthreads();

    if (lane < M_) {
        const int m = lane;
        float a = 0.0f;
        for (int h = 0; h < H_; ++h) a += expf(logits[m][h] - mh[h]) / dh[h];
        attn[m] = a * 0.25f;
        const float ly = params[m * 3 + 0];
        lysh[m] = ly;
        const float ly_n = (ly - 30.0f) / 90.0f;
        const float cu_n = params[m * 3 + 1] / 0.0029f;
        const float ni_n = params[m * 3 + 2] / 0.0018f;
        const float d2 = ((ly_n - t_ly) * (ly_n - t_ly)
                        + (cu_n - t_cu) * (cu_n - t_cu)
                        + (ni_n - t_ni) * (ni_n - t_ni)) / (0.2f * 0.2f);
        swv[m] = expf(-d2 * 0.5f);
    }
    __syncthreads();

    if (lane == 0) {
        float ssum = 0.0f;
        for (int m = 0; m < M_; ++m) ssum += swv[m];
        const float inv1 = 1.0f / (ssum + 1e-12f);
        float wsum = 0.0f;
        for (int m = 0; m < M_; ++m) { wv[m] = attn[m] * (swv[m] * inv1); wsum += wv[m]; }
        const float inv2 = 1.0f / (wsum + 1e-12f);
        for (int m = 0; m < M_; ++m) wv[m] *= inv2;
    }
    __syncthreads();
    if (lane < M_) ws[W_OFF + lane] = wv[lane];

    // ---- ti[t][m] = clip(rint(times[t]/src_t_last[m]*59), 0, 59) ----
    int* wsi = (int*)ws;
    for (int idx = lane; idx < T_ * M_; idx += 32) {
        const int t = idx / M_, m = idx % M_;
        const float tt = (t == T_ - 1) ? 200.0f : (float)t * (200.0f / 49.0f);
        const float r = tt / src_t_last[m] * 59.0f;
        int v = (int)rintf(r);                 // round-to-nearest-even == jnp.round
        v = v < 0 ? 0 : (v > TS_ - 1 ? TS_ - 1 : v);
        wsi[TI_OFF + idx] = v;
    }

    // ---- x axis: ix/fx/mx (searchsorted 'right' - 1, clipped) ----
    auto srcx = [&](int k) -> float {
        return (k == NX_ - 1) ? 100.0f : (float)k * (100.0f / 191.0f);
    };
    for (int i = lane; i < NG_; i += 32) {
        const float qx = (i == NG_ - 1) ? 100.0f : (float)i * (100.0f / 49.0f);
        int idx = 0;
        for (int k = 1; k < NX_; ++k) if (srcx(k) <= qx) idx = k;
        if (idx > NX_ - 2) idx = NX_ - 2;
        const float g0 = srcx(idx), g1 = srcx(idx + 1);
        wsi[IX_OFF + i] = idx;
        ws[FX_OFF + i]  = (qx - g0) / (g1 - g0);
        ws[MX_OFF + i]  = (qx >= 0.0f && qx <= 100.0f) ? 1.0f : 0.0f;
    }

    // ---- y axis per member: stage y_scaled in LDS, then iy/fy/my ----
    for (int idx = lane; idx < M_ * NY_; idx += 32) {
        const int m = idx / NY_;
        ysc[idx] = src_y[idx] * (ly_target / lysh[m]);
    }
    __syncthreads();
    for (int idx = lane; idx < M_ * NG_; idx += 32) {
        const int m = idx / NG_, j = idx % NG_;
        const float qy = ((float)j / 49.0f) * ly_target;
        const float* g = &ysc[m * NY_];
        int iy = 0;
        for (int k = 1; k < NY_; ++k) if (g[k] <= qy) iy = k;
        if (iy > NY_ - 2) iy = NY_ - 2;
        wsi[IY_OFF + idx] = iy;
        ws[FY_OFF + idx]  = (qy - g[iy]) / (g[iy + 1] - g[iy]);
        ws[MY_OFF + idx]  = (qy >= g[0] && qy <= g[NY_ - 1]) ? 1.0f : 0.0f;
    }
}

// One block per (i, t, field); 64 threads cover the j dimension.
// Per thread: 24 members x 4 gathered f32 loads + bilinear + weighted sum.
__global__ __launch_bounds__(64)
void interp_kernel(const float* __restrict__ c1, const float* __restrict__ c2,
                   const float* __restrict__ ws,
                   const float* __restrict__ c_cu_p,
                   const float* __restrict__ c_ni_p,
                   float* __restrict__ out)
{
    const int j = threadIdx.x;
    if (j >= NG_) return;
    const int i = blockIdx.x;
    const int t = blockIdx.y;
    const int f = blockIdx.z;

    const float* __restrict__ preds = (f == 0) ? c1 : c2;
    const int* wsi = (const int*)ws;

    const int   ixi = wsi[IX_OFF + i];
    const float fxi = ws[FX_OFF + i];
    const float mxi = ws[MX_OFF + i];

    float acc = 0.0f;
    #pragma unroll 4
    for (int m = 0; m < M_; ++m) {
        const int tidx = wsi[TI_OFF + t * M_ + m];
        const float* base =
            preds + (((size_t)m * TS_ + (size_t)tidx) * NX_ + (size_t)ixi) * (size_t)NY_;
        const int   iy = wsi[IY_OFF + m * NG_ + j];
        const float fy = ws[FY_OFF + m * NG_ + j];
        const float my = ws[MY_OFF + m * NG_ + j];
        const float v00 = base[iy],        v01 = base[iy + 1];
        const float v10 = base[NY_ + iy],  v11 = base[NY_ + iy + 1];
        const float vx0 = v00 * (1.0f - fxi) + v10 * fxi;
        const float vx1 = v01 * (1.0f - fxi) + v11 * fxi;
        const float v   = vx0 * (1.0f - fy) + vx1 * fy;
        acc += ws[W_OFF + m] * (my * v);
    }
    float res = acc * mxi;
    if (f == 0 && j == 0)       res = c_cu_p[0];
    if (f == 1 && j == NG_ - 1) res = c_ni_p[0];
    out[(((size_t)f * T_ + t) * NG_ + i) * NG_ + j] = res;
}

extern "C" void kernel_launch(void* const* d_in, const int* in_sizes, int n_in,
                              void* d_out, int out_size, void* d_ws, size_t ws_size,
                              hipStream_t stream) {
    const float* c1   = (const float*)d_in[0];   // c1_preds (24,60,192,192)
    const float* c2   = (const float*)d_in[1];   // c2_preds
    const float* prm  = (const float*)d_in[2];   // params (24,3)
    const float* sy   = (const float*)d_in[3];   // src_y (24,192)
    const float* stl  = (const float*)d_in[4];   // src_t_last (24,)
    const float* Wq   = (const float*)d_in[5];   // (3,32)
    const float* bq   = (const float*)d_in[6];   // (32,)
    const float* Wk   = (const float*)d_in[7];   // (3,32)
    const float* bk   = (const float*)d_in[8];   // (32,)
    const float* lyt  = (const float*)d_in[9];   // ly_target scalar
    const float* ccu  = (const float*)d_in[10];  // c_cu_target scalar
    const float* cni  = (const float*)d_in[11];  // c_ni_target scalar
    float* ws = (float*)d_ws;

    setup_kernel<<<1, 32, 0, stream>>>(prm, sy, stl, Wq, bq, Wk, bk, lyt, ccu, cni, ws);
    interp_kernel<<<dim3(NG_, T_, 2), 64, 0, stream>>>(c1, c2, ws, ccu, cni, (float*)d_out);
}